// FNO_4698694222627
// MI455X (gfx1250) — compile-verified
//
#include <hip/hip_runtime.h>

// ---------------------------------------------------------------------------
// FNO forward for MI455X (gfx1250, wave32, WMMA).
// All heavy math is f32 WMMA 16x16x4 GEMMs; spectral transforms are partial
// DFTs (only 32 retained modes per axis) expressed as GEMMs vs twiddle tables.
// GEMM uses double-buffered LDS with fragment-ready B layout (single
// ds_load_b64 per B fragment, no repacking moves) and register-staged
// global prefetch overlapped with the WMMA stream.
// ---------------------------------------------------------------------------

typedef float v2f __attribute__((ext_vector_type(2)));
typedef float v8f __attribute__((ext_vector_type(8)));

__device__ __forceinline__ float gelu_f(float x) {
    return 0.5f * x * (1.0f + erff(x * 0.70710678118654752f));
}

// ---------------------------------------------------------------------------
// Generic f32 WMMA GEMM:  D[M,N] = A[M,K] * B[K,N]  (+bias[m]) (+resid) (gelu)
// Block = 128 threads = 4 waves. Each wave owns a 16-row M sub-tile; the block
// covers 64 M rows x (16*NT) N cols. B chunks [16 x BN] staged through LDS in
// fragment-ready row-pair layout, double buffered. Requires: M % 64 == 0,
// N % (16*NT) == 0, K % 16 == 0. Batched via blockIdx.z with per-operand
// batch strides (0 => operand shared across the batch).
// ---------------------------------------------------------------------------
template <int NT>
__global__ __launch_bounds__(128) void gemm_wmma_f32(
    const float* __restrict__ A, const float* __restrict__ B,
    float* __restrict__ D, const float* __restrict__ bias,
    const float* __restrict__ resid, int gelu,
    int K, int lda, int ldb, int ldd,
    long long sA, long long sB, long long sD) {
    constexpr int BN  = 16 * NT;      // N cols per block
    constexpr int LRS = 2 * BN + 32;  // LDS row-pair stride (floats); +32 puts the
                                      // second lane-half on banks 32..63 (b64 reads)
    constexpr int CH  = (16 * BN) / 128;  // B elements per thread per chunk
    __shared__ float lb[2][8][LRS];   // [buffer][row-pair][col*2 + half]

    const int tid  = threadIdx.x;
    const int wave = tid >> 5;
    const int lane = tid & 31;
    const long long z = blockIdx.z;

    const float* Ab = A + z * sA;
    const float* Bb = B + z * sB;
    float*       Db = D + z * sD;

    const int n0   = blockIdx.x * BN;
    const int m0   = blockIdx.y * 64 + wave * 16;
    const int mrow = lane & 15;
    const int kb   = (lane >> 4) << 1;  // 0 for lanes 0-15, 2 for lanes 16-31
    const int ncol = lane & 15;
    const int hi   = lane >> 4;

    v8f acc[NT];
#pragma unroll
    for (int t = 0; t < NT; ++t)
#pragma unroll
        for (int e = 0; e < 8; ++e) acc[t][e] = 0.0f;

    float rbuf[CH];
    // Prefetch chunk 0 into registers, stage into LDS buffer 0.
#pragma unroll
    for (int j = 0; j < CH; ++j) {
        int i = tid + j * 128, r = i / BN, c = i - r * BN;
        rbuf[j] = Bb[(long long)r * ldb + (n0 + c)];
    }
#pragma unroll
    for (int j = 0; j < CH; ++j) {
        int i = tid + j * 128, r = i / BN, c = i - r * BN;
        lb[0][r >> 1][2 * c + (r & 1)] = rbuf[j];
    }
    __syncthreads();

    const int nchunks = K >> 4;
    for (int ch = 0; ch < nchunks; ++ch) {
        const int cur = ch & 1;
        const int kc  = ch << 4;
        // Register-staged prefetch of the next chunk (overlaps the WMMAs).
        if (ch + 1 < nchunks) {
#pragma unroll
            for (int j = 0; j < CH; ++j) {
                int i = tid + j * 128, r = i / BN, c = i - r * BN;
                rbuf[j] = Bb[(long long)(kc + 16 + r) * ldb + (n0 + c)];
            }
        }
#pragma unroll
        for (int kk = 0; kk < 4; ++kk) {
            // A fragment (16x4 f32): lane = M row; VGPR pair = K {kb, kb+1}
            const float* ap = Ab + (long long)(m0 + mrow) * lda + (kc + kk * 4 + kb);
            v2f a;
            a.x = ap[0];
            a.y = ap[1];
            const int rp = kk * 2 + hi;  // row-pair holding K rows {kb, kb+1}
#pragma unroll
            for (int t = 0; t < NT; ++t) {
                v2f b = *(const v2f*)&lb[cur][rp][2 * (t * 16 + ncol)];
                acc[t] = __builtin_amdgcn_wmma_f32_16x16x4_f32(
                    false, a, false, b, (short)0, acc[t], false, false);
            }
        }
        if (ch + 1 < nchunks) {
#pragma unroll
            for (int j = 0; j < CH; ++j) {
                int i = tid + j * 128, r = i / BN, c = i - r * BN;
                lb[cur ^ 1][r >> 1][2 * c + (r & 1)] = rbuf[j];
            }
            __syncthreads();
        }
    }

    // Epilogue. C/D layout: acc element r -> M = r + 8*(lane>=16), N = lane&15.
    const int mhi = hi * 8;
#pragma unroll
    for (int t = 0; t < NT; ++t) {
#pragma unroll
        for (int r = 0; r < 8; ++r) {
            int m = m0 + r + mhi;
            int n = n0 + t * 16 + ncol;
            float v = acc[t][r];
            if (bias)  v += bias[m];
            if (resid) v += resid[z * sD + (long long)m * ldd + n];
            if (gelu)  v = gelu_f(v);
            Db[(long long)m * ldd + n] = v;
        }
    }
}

// ---------------------------------------------------------------------------
// Lifting conv (CIN=1): h[b,c,pix] = p_w[c]*x[b,pix] + p_b[c]
// ---------------------------------------------------------------------------
__global__ __launch_bounds__(256) void lift_kernel(
    const float* __restrict__ x, const float* __restrict__ pw,
    const float* __restrict__ pb, float* __restrict__ h) {
    long long idx = (long long)blockIdx.x * 256 + threadIdx.x;  // [b(16)][c(64)][pix(65536)]
    int c = (int)((idx >> 16) & 63);
    long long xi = ((idx >> 22) << 16) | (idx & 65535);
    h[idx] = pw[c] * x[xi] + pb[c];
}

// ---------------------------------------------------------------------------
// Twiddle tables (built once per launch, deterministic):
//  DFTW [256][64]: forward W-rDFT,   cols 0..31 = cos, 32..63 = -sin
//  T2   [128][512]: forward H-DFT (complex, rows=(j,part), cols=(h,part))
//  Ti   [512][128]: inverse H-DFT * (1/256)
//  G    [64][256]:  inverse W-irfft * (1/256), Hermitian weighting (2x for k>0)
// ---------------------------------------------------------------------------
__global__ __launch_bounds__(256) void init_tables(
    float* __restrict__ DFTW, float* __restrict__ T2,
    float* __restrict__ Ti, float* __restrict__ G) {
    int idx = blockIdx.x * 256 + threadIdx.x;
    const float w0 = 6.283185307179586f / 256.0f;
    if (idx < 16384) {  // DFTW
        int n = idx >> 6, k = idx & 63, kk = k & 31;
        float ang = w0 * (float)((kk * n) & 255);
        DFTW[idx] = (k < 32) ? cosf(ang) : -sinf(ang);
    } else if (idx < 16384 + 65536) {  // T2
        int i2 = idx - 16384;
        int m = i2 >> 9, col = i2 & 511;
        int j = m >> 1, p = m & 1;
        int hh = col >> 1, q = col & 1;
        int k1 = (j < 32) ? j : (192 + j);  // modes 0..31 and 224..255
        float ang = w0 * (float)((k1 * hh) & 255);
        float c = cosf(ang), s = sinf(ang);
        T2[i2] = p == 0 ? (q == 0 ? c : s) : (q == 0 ? -s : c);
    } else if (idx < 16384 + 65536 + 65536) {  // Ti
        int i2 = idx - (16384 + 65536);
        int m = i2 >> 7, col = i2 & 127;
        int hh = m >> 1, p = m & 1;
        int j = col >> 1, q = col & 1;
        int k1 = (j < 32) ? j : (192 + j);
        float ang = w0 * (float)((k1 * hh) & 255);
        float c = cosf(ang), s = sinf(ang);
        float v = p == 0 ? (q == 0 ? c : -s) : (q == 0 ? s : c);
        Ti[i2] = v * (1.0f / 256.0f);
    } else if (idx < 163840) {  // G
        int i2 = idx - (16384 + 65536 + 65536);
        int k = i2 >> 8, w = i2 & 255, kk = k & 31;
        float ang = w0 * (float)((kk * w) & 255);
        float v;
        if (k < 32) v = ((kk == 0) ? 1.0f : 2.0f) * cosf(ang) * (1.0f / 256.0f);
        else        v = (kk == 0) ? 0.0f : (-2.0f / 256.0f) * sinf(ang);
        G[i2] = v;
    }
}

// ---------------------------------------------------------------------------
// Per-mode complex channel mixing: Z[b,o,(j,p),k2] = sum_i Y2[b,i,(j,p),k2]*W
// Y2/Z layout: [b][ch][(k1idx,part)=128][k2=32]. Weights [i][o][m1][m2].
// ---------------------------------------------------------------------------
__global__ __launch_bounds__(256) void specmul_kernel(
    const float* __restrict__ Y2,
    const float* __restrict__ w1r, const float* __restrict__ w1i,
    const float* __restrict__ w2r, const float* __restrict__ w2i,
    float* __restrict__ Z) {
    int k2 = threadIdx.x & 31;
    int o  = (threadIdx.x >> 5) + blockIdx.x * 8;
    int j  = blockIdx.y;
    int b  = blockIdx.z;
    const float* wr = (j < 32) ? w1r : w2r;
    const float* wi = (j < 32) ? w1i : w2i;
    int jm = (j < 32) ? j : (j - 32);

    const float* Yb = Y2 + (long long)b * 64 * 4096 + (2 * j) * 32 + k2;
    long long wbase = ((long long)o * 32 + jm) * 32 + k2;
    float zr = 0.0f, zi = 0.0f;
    for (int i = 0; i < 64; ++i) {
        float yr = Yb[(long long)i * 4096];
        float yi = Yb[(long long)i * 4096 + 32];
        float a = wr[wbase + (long long)i * 65536];
        float c = wi[wbase + (long long)i * 65536];
        zr += yr * a - yi * c;
        zi += yr * c + yi * a;
    }
    float* Zp = Z + ((long long)b * 64 + o) * 4096 + (2 * j) * 32 + k2;
    Zp[0]  = zr;
    Zp[32] = zi;
}

// ---------------------------------------------------------------------------
// Fused projection head: out = q2w . gelu(q1w . h + q1b) + q2b, per pixel.
// q1w tiled through LDS in 64-row chunks; h channels held in registers.
// ---------------------------------------------------------------------------
__global__ __launch_bounds__(256) void proj_kernel(
    const float* __restrict__ h, const float* __restrict__ q1w,
    const float* __restrict__ q1b, const float* __restrict__ q2w,
    const float* __restrict__ q2b, float* __restrict__ out) {
    __shared__ float w1s[64 * 64];
    int pix = blockIdx.x * 256 + threadIdx.x;
    int b   = blockIdx.y;
    const float* hp = h + (long long)b * 64 * 65536 + pix;
    float hv[64];
#pragma unroll
    for (int i = 0; i < 64; ++i) hv[i] = hp[(long long)i * 65536];
    float acc = q2b[0];
    for (int jc = 0; jc < 256; jc += 64) {
        __syncthreads();
        for (int i = threadIdx.x; i < 4096; i += 256) w1s[i] = q1w[jc * 64 + i];
        __syncthreads();
        for (int jj = 0; jj < 64; ++jj) {
            int j = jc + jj;
            float s = q1b[j];
#pragma unroll
            for (int i = 0; i < 64; ++i) s += w1s[jj * 64 + i] * hv[i];
            acc += q2w[j] * gelu_f(s);
        }
    }
    out[(long long)b * 65536 + pix] = acc;
}

// ---------------------------------------------------------------------------
extern "C" void kernel_launch(void* const* d_in, const int* in_sizes, int n_in,
                              void* d_out, int out_size, void* d_ws, size_t ws_size,
                              hipStream_t stream) {
    const float* x    = (const float*)d_in[0];
    const float* p_w  = (const float*)d_in[1];
    const float* p_b  = (const float*)d_in[2];
    const float* sw1r = (const float*)d_in[3];
    const float* sw1i = (const float*)d_in[4];
    const float* sw2r = (const float*)d_in[5];
    const float* sw2i = (const float*)d_in[6];
    const float* m1w  = (const float*)d_in[7];
    const float* m1b  = (const float*)d_in[8];
    const float* m2w  = (const float*)d_in[9];
    const float* m2b  = (const float*)d_in[10];
    const float* skw  = (const float*)d_in[11];
    const float* skb  = (const float*)d_in[12];
    const float* q1w  = (const float*)d_in[13];
    const float* q1b  = (const float*)d_in[14];
    const float* q2w  = (const float*)d_in[15];
    const float* q2b  = (const float*)d_in[16];

    float* ws = (float*)d_ws;
    size_t off = 0;
    auto carve = [&](size_t n) { float* p = ws + off; off += n; return p; };
    float* big0 = carve(67108864);  // [16][64][65536]
    float* big1 = carve(67108864);
    float* big2 = carve(67108864);
    float* Y1   = carve(16777216);  // [16][64][256][64]
    float* Y2   = carve(4194304);   // [16][64][128][32]
    float* Zb   = carve(4194304);   // [16][64][128][32]
    float* IH   = carve(16777216);  // [16][64][512][32]
    float* DFTW = carve(16384);
    float* T2   = carve(65536);
    float* Ti   = carve(65536);
    float* G    = carve(16384);

    init_tables<<<640, 256, 0, stream>>>(DFTW, T2, Ti, G);
    lift_kernel<<<262144, 256, 0, stream>>>(x, p_w, p_b, big0);

    float* h  = big0;
    float* p1 = big1;
    float* p2 = big2;
    for (int l = 0; l < 4; ++l) {
        // Forward W-rDFT: rows=(b,c,hrow), K=256(w), N=64 comps
        gemm_wmma_f32<4><<<dim3(1, 4096, 1), 128, 0, stream>>>(
            h, DFTW, Y1, nullptr, nullptr, 0, 256, 256, 64, 64, 0, 0, 0);
        // Forward H-DFT: per (b,c): M=128, K=512, N=32
        gemm_wmma_f32<2><<<dim3(1, 2, 1024), 128, 0, stream>>>(
            T2, Y1, Y2, nullptr, nullptr, 0, 512, 512, 32, 32, 0, 16384, 4096);
        // Complex channel mixing on retained modes
        specmul_kernel<<<dim3(8, 64, 16), 256, 0, stream>>>(
            Y2, sw1r + (long long)l * 4194304, sw1i + (long long)l * 4194304,
            sw2r + (long long)l * 4194304, sw2i + (long long)l * 4194304, Zb);
        // Inverse H transform: per (b,o): M=512, K=128, N=32
        gemm_wmma_f32<2><<<dim3(1, 8, 1024), 128, 0, stream>>>(
            Ti, Zb, IH, nullptr, nullptr, 0, 128, 128, 32, 32, 0, 4096, 16384);
        // Inverse W irfft: rows=(b,o,hrow), K=64, N=256(w) -> spatial x1
        gemm_wmma_f32<4><<<dim3(4, 4096, 1), 128, 0, stream>>>(
            IH, G, p1, nullptr, nullptr, 0, 64, 64, 256, 256, 0, 0, 0);
        // MLP conv1 + GELU
        gemm_wmma_f32<4><<<dim3(1024, 1, 16), 128, 0, stream>>>(
            m1w + l * 4096, p1, p2, m1b + l * 64, nullptr, 1,
            64, 64, 65536, 65536, 0, 4194304, 4194304);
        // MLP conv2
        gemm_wmma_f32<4><<<dim3(1024, 1, 16), 128, 0, stream>>>(
            m2w + l * 4096, p2, p1, m2b + l * 64, nullptr, 0,
            64, 64, 65536, 65536, 0, 4194304, 4194304);
        // Skip conv + residual add (+GELU except last layer)
        gemm_wmma_f32<4><<<dim3(1024, 1, 16), 128, 0, stream>>>(
            skw + l * 4096, h, p2, skb + l * 64, p1, (l < 3) ? 1 : 0,
            64, 64, 65536, 65536, 0, 4194304, 4194304);
        float* nh = p2;  // rotate: new h = p2; free = {old h, p1}
        p2 = p1;
        p1 = h;
        h  = nh;
    }
    proj_kernel<<<dim3(256, 16), 256, 0, stream>>>(h, q1w, q1b, q2w, q2b,
                                                   (float*)d_out);
}